// GalaxyLSSBackbone_69088843924129
// MI455X (gfx1250) — compile-verified
//
#include <hip/hip_runtime.h>
#include <math.h>

#define NPTS  4096
#define HID   64
#define EPSF  1e-8f
#define SPLIT 8                    // i-dimension split for pairwise passes
#define CHUNK (NPTS / SPLIT)       // 512 sources per block

typedef __attribute__((ext_vector_type(2))) float v2f;
typedef __attribute__((ext_vector_type(8))) float v8f;

// ---------------------------------------------------------------------------
// Pass 1 (split-i): partial sums over i in [y*CHUNK, (y+1)*CHUNK):
//   q0=Σw, q1=Σw·a, q2=Σw·b, q3=Σw·cos2φ, q4=Σw·sin2φ
// Partial layout: part[(q*SPLIT + y)*NPTS + j]
// ---------------------------------------------------------------------------
__global__ __launch_bounds__(256) void pass1_kernel(
    const float* __restrict__ pos, const float* __restrict__ shapes,
    float* __restrict__ part)
{
    __shared__ float4 tile[256];
    const int j  = blockIdx.x * 256 + threadIdx.x;
    const int y  = blockIdx.y;
    const int i0 = y * CHUNK;
    const float2 pj = *(const float2*)(pos + 2 * j);
    float s0 = 0.f, s1 = 0.f, s2 = 0.f, s3 = 0.f, s4 = 0.f;

    for (int t = i0; t < i0 + CHUNK; t += 256) {
        const int i = t + threadIdx.x;
        const float2 pi = *(const float2*)(pos + 2 * i);
        const float2 si = *(const float2*)(shapes + 2 * i);
        tile[threadIdx.x] = make_float4(pi.x, pi.y, si.x, si.y);
        if (t + 256 < i0 + CHUNK)
            __builtin_prefetch(pos + 2 * (i + 256), 0, 1);
        __syncthreads();
        #pragma unroll 8
        for (int k = 0; k < 256; ++k) {
            const float4 T = tile[k];
            const float dx = pj.x - T.x, dy = pj.y - T.y;  // delta = pos[j]-pos[i]
            const float r2 = dx * dx + dy * dy;
            float w   = 1.0f / (sqrtf(r2 + EPSF) + EPSF);
            float inv = 1.0f / r2;
            if (t + k == j) { w = 0.f; inv = 0.f; }        // kill diagonal
            const float c2p = (dx * dx - dy * dy) * inv;   // cos(2*atan2(dy,dx))
            const float s2p = (2.f * dx * dy) * inv;       // sin(2*atan2(dy,dx))
            s0 += w;
            s1 += w * T.z;
            s2 += w * T.w;
            s3 += w * c2p;
            s4 += w * s2p;
        }
        __syncthreads();
    }
    part[(0 * SPLIT + y) * NPTS + j] = s0;
    part[(1 * SPLIT + y) * NPTS + j] = s1;
    part[(2 * SPLIT + y) * NPTS + j] = s2;
    part[(3 * SPLIT + y) * NPTS + j] = s3;
    part[(4 * SPLIT + y) * NPTS + j] = s4;
}

// ---------------------------------------------------------------------------
// Reduce pass-1 partials (fixed order => deterministic) + per-point finalize:
// alpha, c=cos2a, s=sin2a, rel_cos/rel_sin, p/q, feature slots 0,3,4 (+pad).
// ---------------------------------------------------------------------------
__global__ __launch_bounds__(256) void finalize1_kernel(
    const float* __restrict__ shapes, const float* __restrict__ redshift,
    const float* __restrict__ part,
    float* __restrict__ WS, float* __restrict__ C, float* __restrict__ S,
    float* __restrict__ P, float* __restrict__ Q,
    float* __restrict__ feat8)
{
    const int j = blockIdx.x * 256 + threadIdx.x;
    float S1 = 0.f, S2 = 0.f, S3 = 0.f, S4 = 0.f, S5 = 0.f;
    #pragma unroll
    for (int y = 0; y < SPLIT; ++y) {
        S1 += part[(0 * SPLIT + y) * NPTS + j];
        S2 += part[(1 * SPLIT + y) * NPTS + j];
        S3 += part[(2 * SPLIT + y) * NPTS + j];
        S4 += part[(3 * SPLIT + y) * NPTS + j];
        S5 += part[(4 * SPLIT + y) * NPTS + j];
    }
    const float m2 = S2 * S2 + S3 * S3;
    float alpha = 0.5f * atan2f(S3, S2);
    if (m2 < EPSF) alpha = 0.f;
    const float c = cosf(2.f * alpha);
    const float s = sinf(2.f * alpha);
    const float wsum = fmaxf(S1, EPSF);
    const float inv_ws = 1.0f / wsum;

    const float rel_cos = (c * S4 + s * S5) * inv_ws;
    const float rel_sin = (c * S5 - s * S4) * inv_ws;

    const float a = shapes[2 * j], b = shapes[2 * j + 1];
    WS[j] = wsum;
    C[j] = c; S[j] = s;
    P[j] = a * c - b * s;
    Q[j] = a * s + b * c;

    feat8[j * 8 + 0] = redshift[j];
    feat8[j * 8 + 3] = rel_cos;
    feat8[j * 8 + 4] = rel_sin;
    feat8[j * 8 + 5] = 0.f;
    feat8[j * 8 + 6] = 0.f;
    feat8[j * 8 + 7] = 0.f;
}

// ---------------------------------------------------------------------------
// Pass 2 (split-i): partials of T1=Σw·p_i, T2=Σw·q_i
// Partial layout: part[(q*SPLIT + y)*NPTS + j], q in {0,1} (region reused)
// ---------------------------------------------------------------------------
__global__ __launch_bounds__(256) void pass2_kernel(
    const float* __restrict__ pos,
    const float* __restrict__ P, const float* __restrict__ Q,
    float* __restrict__ part)
{
    __shared__ float4 tile[256];
    const int j  = blockIdx.x * 256 + threadIdx.x;
    const int y  = blockIdx.y;
    const int i0 = y * CHUNK;
    const float2 pj = *(const float2*)(pos + 2 * j);
    float t1 = 0.f, t2 = 0.f;

    for (int t = i0; t < i0 + CHUNK; t += 256) {
        const int i = t + threadIdx.x;
        const float2 pi = *(const float2*)(pos + 2 * i);
        tile[threadIdx.x] = make_float4(pi.x, pi.y, P[i], Q[i]);
        if (t + 256 < i0 + CHUNK)
            __builtin_prefetch(pos + 2 * (i + 256), 0, 1);
        __syncthreads();
        #pragma unroll 8
        for (int k = 0; k < 256; ++k) {
            const float4 T = tile[k];
            const float dx = pj.x - T.x, dy = pj.y - T.y;
            const float r2 = dx * dx + dy * dy;
            float w = 1.0f / (sqrtf(r2 + EPSF) + EPSF);
            if (t + k == j) w = 0.f;
            t1 += w * T.z;
            t2 += w * T.w;
        }
        __syncthreads();
    }
    part[(0 * SPLIT + y) * NPTS + j] = t1;
    part[(1 * SPLIT + y) * NPTS + j] = t2;
}

// ---------------------------------------------------------------------------
// Reduce pass-2 partials + write feature slots 1,2 (rot_cos2 / rot_sin2).
// ---------------------------------------------------------------------------
__global__ __launch_bounds__(256) void finalize2_kernel(
    const float* __restrict__ part,
    const float* __restrict__ WS,
    const float* __restrict__ C, const float* __restrict__ S,
    float* __restrict__ feat8)
{
    const int j = blockIdx.x * 256 + threadIdx.x;
    float t1 = 0.f, t2 = 0.f;
    #pragma unroll
    for (int y = 0; y < SPLIT; ++y) {
        t1 += part[(0 * SPLIT + y) * NPTS + j];
        t2 += part[(1 * SPLIT + y) * NPTS + j];
    }
    const float inv_ws = 1.0f / WS[j];
    const float cj = C[j], sj = S[j];
    feat8[j * 8 + 1] = (cj * t1 + sj * t2) * inv_ws;   // rot_cos2
    feat8[j * 8 + 2] = (cj * t2 - sj * t1) * inv_ws;   // rot_sin2
}

// ---------------------------------------------------------------------------
// Fused 3-layer MLP with V_WMMA_F32_16X16X4_F32 (unchanged: 136 wmma/wave).
// 128 threads = 4 waves; each wave owns 16 rows; workgroup owns 64 rows.
// ---------------------------------------------------------------------------
__global__ __launch_bounds__(128) void mlp3_kernel(
    const float* __restrict__ feat8,
    const float* __restrict__ W1, const float* __restrict__ b1,
    const float* __restrict__ W2, const float* __restrict__ b2,
    const float* __restrict__ W3, const float* __restrict__ b3,
    float* __restrict__ out)
{
    __shared__ float sW1[8 * HID];       // W1 zero-padded 5->8 rows (K)
    __shared__ float sW[HID * HID];      // W2, then W3
    __shared__ float sB[3 * HID];
    __shared__ float hA[64 * HID];       // layer-1 activations (64 rows)
    __shared__ float hB[64 * HID];       // layer-2 activations

    const int tid  = threadIdx.x;
    const int wave = tid >> 5;           // 0..3
    const int lane = tid & 31;
    const int lo   = lane & 15;          // row (A) / col (B,D) index
    const int hi   = lane >> 4;          // K-half selector
    const int rowBase = blockIdx.x * 64;

    for (int idx = tid; idx < 8 * HID; idx += 128) {
        const int k = idx / HID;
        sW1[idx] = (k < 5) ? W1[idx] : 0.0f;
    }
    for (int idx = tid; idx < HID * HID; idx += 128) sW[idx] = W2[idx];
    if (tid < HID) {
        sB[tid]           = b1[tid];
        sB[HID + tid]     = b2[tid];
        sB[2 * HID + tid] = b3[tid];
    }
    __syncthreads();

    // ---- layer 1: feat8 (K=8) x W1 -> relu -> hA
    {
        v8f acc[4] = {};
        const int r0 = rowBase + wave * 16;
        #pragma unroll
        for (int kk = 0; kk < 8; kk += 4) {
            const float* ap = feat8 + (size_t)(r0 + lo) * 8 + kk + 2 * hi;
            v2f a; a.x = ap[0]; a.y = ap[1];
            #pragma unroll
            for (int nt = 0; nt < 4; ++nt) {
                v2f bf;
                bf.x = sW1[(kk + 2 * hi + 0) * HID + nt * 16 + lo];
                bf.y = sW1[(kk + 2 * hi + 1) * HID + nt * 16 + lo];
                acc[nt] = __builtin_amdgcn_wmma_f32_16x16x4_f32(
                    false, a, false, bf, (short)0, acc[nt], false, false);
            }
        }
        #pragma unroll
        for (int nt = 0; nt < 4; ++nt) {
            const int col = nt * 16 + lo;
            const float bias = sB[col];
            #pragma unroll
            for (int v = 0; v < 8; ++v) {
                const int row = wave * 16 + v + 8 * hi;
                hA[row * HID + col] = fmaxf(acc[nt][v] + bias, 0.0f);
            }
        }
    }
    __syncthreads();

    // ---- layer 2: hA (K=64) x W2 -> relu -> hB
    {
        v8f acc[4] = {};
        #pragma unroll
        for (int kk = 0; kk < HID; kk += 4) {
            const v2f a = *(const v2f*)&hA[(wave * 16 + lo) * HID + kk + 2 * hi];
            #pragma unroll
            for (int nt = 0; nt < 4; ++nt) {
                v2f bf;
                bf.x = sW[(kk + 2 * hi + 0) * HID + nt * 16 + lo];
                bf.y = sW[(kk + 2 * hi + 1) * HID + nt * 16 + lo];
                acc[nt] = __builtin_amdgcn_wmma_f32_16x16x4_f32(
                    false, a, false, bf, (short)0, acc[nt], false, false);
            }
        }
        #pragma unroll
        for (int nt = 0; nt < 4; ++nt) {
            const int col = nt * 16 + lo;
            const float bias = sB[HID + col];
            #pragma unroll
            for (int v = 0; v < 8; ++v) {
                const int row = wave * 16 + v + 8 * hi;
                hB[row * HID + col] = fmaxf(acc[nt][v] + bias, 0.0f);
            }
        }
    }
    __syncthreads();

    for (int idx = tid; idx < HID * HID; idx += 128) sW[idx] = W3[idx];
    __syncthreads();

    // ---- layer 3: hB (K=64) x W3 -> out
    {
        v8f acc[4] = {};
        #pragma unroll
        for (int kk = 0; kk < HID; kk += 4) {
            const v2f a = *(const v2f*)&hB[(wave * 16 + lo) * HID + kk + 2 * hi];
            #pragma unroll
            for (int nt = 0; nt < 4; ++nt) {
                v2f bf;
                bf.x = sW[(kk + 2 * hi + 0) * HID + nt * 16 + lo];
                bf.y = sW[(kk + 2 * hi + 1) * HID + nt * 16 + lo];
                acc[nt] = __builtin_amdgcn_wmma_f32_16x16x4_f32(
                    false, a, false, bf, (short)0, acc[nt], false, false);
            }
        }
        #pragma unroll
        for (int nt = 0; nt < 4; ++nt) {
            const int col = nt * 16 + lo;
            const float bias = sB[2 * HID + col];
            #pragma unroll
            for (int v = 0; v < 8; ++v) {
                const int row = rowBase + wave * 16 + v + 8 * hi;
                out[(size_t)row * HID + col] = acc[nt][v] + bias;
            }
        }
    }
}

// ---------------------------------------------------------------------------
extern "C" void kernel_launch(void* const* d_in, const int* in_sizes, int n_in,
                              void* d_out, int out_size, void* d_ws, size_t ws_size,
                              hipStream_t stream) {
    const float* pos      = (const float*)d_in[0];
    const float* redshift = (const float*)d_in[1];
    const float* shapes   = (const float*)d_in[2];
    const float* W1       = (const float*)d_in[3];
    const float* b1       = (const float*)d_in[4];
    const float* W2       = (const float*)d_in[5];
    const float* b2       = (const float*)d_in[6];
    const float* W3       = (const float*)d_in[7];
    const float* b3       = (const float*)d_in[8];

    float* ws = (float*)d_ws;
    float* WS    = ws + 0 * NPTS;        // clipped weight sums
    float* C     = ws + 1 * NPTS;
    float* S     = ws + 2 * NPTS;
    float* P     = ws + 3 * NPTS;
    float* Q     = ws + 4 * NPTS;
    float* feat8 = ws + 5 * NPTS;        // NPTS x 8, K-padded
    float* part  = ws + 13 * NPTS;       // 5*SPLIT*NPTS partials (reused by pass 2)

    dim3 pgrid(NPTS / 256, SPLIT);
    pass1_kernel<<<pgrid, 256, 0, stream>>>(pos, shapes, part);
    finalize1_kernel<<<NPTS / 256, 256, 0, stream>>>(shapes, redshift, part,
                                                     WS, C, S, P, Q, feat8);
    pass2_kernel<<<pgrid, 256, 0, stream>>>(pos, P, Q, part);
    finalize2_kernel<<<NPTS / 256, 256, 0, stream>>>(part, WS, C, S, feat8);
    mlp3_kernel<<<NPTS / 64, 128, 0, stream>>>(feat8, W1, b1, W2, b2, W3, b3,
                                               (float*)d_out);
}